// GraphPoolingModel_layer_1_51616916963374
// MI455X (gfx1250) — compile-verified
//
#include <hip/hip_runtime.h>
#include <hip/hip_bf16.h>

typedef __attribute__((ext_vector_type(2))) float v2f;
typedef __attribute__((ext_vector_type(8))) float v8f;

#define Dd 128
#define Hh 128
#define Gg 64

// ---------------------------------------------------------------------------
// Zero-init scratch
// ---------------------------------------------------------------------------
__global__ void zero_ws_kernel(float* __restrict__ p, size_t cnt) {
    size_t i = (size_t)blockIdx.x * blockDim.x + threadIdx.x;
    if (i < cnt) p[i] = 0.0f;
}

// ---------------------------------------------------------------------------
// h = x @ W_gcn  via V_WMMA_F32_16X16X4_F32
// grid.x = ceil(N/16) row tiles; block = 256 (8 waves), wave w owns cols
// [16w, 16w+16). A 16x4 f32 layout: lanes 0-15 -> K = k+0,k+1 ; lanes 16-31
// -> K = k+2,k+3 (M = lane&15 in both halves). B mirrored. C/D: VGPR r ->
// M = r + 8*(lane>=16), N = lane&15.
// Epilogue: uniform fast path (full tile) -> 8 unguarded coalesced stores;
// ragged last tile only takes the guarded path.
// ---------------------------------------------------------------------------
__global__ void gemm_xw_wmma(const float* __restrict__ X,
                             const float* __restrict__ W,
                             float* __restrict__ Hout, int n) {
    const int wave  = threadIdx.x >> 5;
    const int lane  = threadIdx.x & 31;
    const int l15   = lane & 15;
    const int lhalf = lane >> 4;          // 0 or 1
    const int m0    = blockIdx.x * 16;
    const int n0    = wave * 16;

    int arow = m0 + l15;
    if (arow >= n) arow = n - 1;          // clamp loads; stores handled below
    const float* __restrict__ xrow = X + (size_t)arow * Dd;
    const int ka = lhalf * 2;             // K sub-offset for this lane half

    v8f acc = {};
    #pragma unroll 4
    for (int k = 0; k < Dd; k += 4) {
        v2f a, b;
        a.x = xrow[k + ka + 0];
        a.y = xrow[k + ka + 1];
        b.x = W[(size_t)(k + ka + 0) * Hh + n0 + l15];
        b.y = W[(size_t)(k + ka + 1) * Hh + n0 + l15];
        acc = __builtin_amdgcn_wmma_f32_16x16x4_f32(
            /*neg_a=*/false, a, /*neg_b=*/false, b,
            /*c_mod=*/(short)0, acc, /*reuse_a=*/false, /*reuse_b=*/false);
    }

    float* __restrict__ orow =
        Hout + (size_t)(m0 + lhalf * 8) * Hh + n0 + l15;
    if (m0 + 16 <= n) {
        // full tile: unguarded, coalesced b32 stores (16 lanes per row)
        #pragma unroll
        for (int r = 0; r < 8; ++r) orow[(size_t)r * Hh] = acc[r];
    } else {
        #pragma unroll
        for (int r = 0; r < 8; ++r) {
            if (m0 + r + lhalf * 8 < n) orow[(size_t)r * Hh] = acc[r];
        }
    }
}

// ---------------------------------------------------------------------------
// deg[dst] += 1 per edge
// ---------------------------------------------------------------------------
__global__ void deg_count_kernel(const int* __restrict__ dst,
                                 float* __restrict__ deg, int e) {
    int i = blockIdx.x * blockDim.x + threadIdx.x;
    if (i < e) atomicAdd(&deg[dst[i]], 1.0f);
}

// ---------------------------------------------------------------------------
// deg -> rsqrt(deg+1) in place; counts[batch[i]] += 1
// ---------------------------------------------------------------------------
__global__ void dinv_counts_kernel(float* __restrict__ deg,
                                   const int* __restrict__ batch,
                                   float* __restrict__ counts, int n) {
    int i = blockIdx.x * blockDim.x + threadIdx.x;
    if (i < n) {
        deg[i] = rsqrtf(deg[i] + 1.0f);
        atomicAdd(&counts[batch[i]], 1.0f);
    }
}

// ---------------------------------------------------------------------------
// Per-edge scatter: agg[dst] += h[src] * dinv[src]*dinv[dst]
// One wave32 per edge, float4 per lane (32*4 = 128 = H). L2-resident atomics.
// ---------------------------------------------------------------------------
__global__ void edge_scatter_kernel(const int* __restrict__ src,
                                    const int* __restrict__ dst,
                                    const float* __restrict__ Hlin,
                                    const float* __restrict__ dinv,
                                    float* __restrict__ Agg, int e) {
    int widx = blockIdx.x * (blockDim.x >> 5) + (threadIdx.x >> 5);
    int lane = threadIdx.x & 31;
    if (widx >= e) return;
    int s = src[widx];
    int d = dst[widx];
    float coef = dinv[s] * dinv[d];
    const float4* __restrict__ hp = (const float4*)(Hlin + (size_t)s * Hh);
    float4 hv = hp[lane];
    float* __restrict__ ap = Agg + (size_t)d * Hh + lane * 4;
    atomicAdd(ap + 0, hv.x * coef);
    atomicAdd(ap + 1, hv.y * coef);
    atomicAdd(ap + 2, hv.z * coef);
    atomicAdd(ap + 3, hv.w * coef);
}

// ---------------------------------------------------------------------------
// Fused: self-loop + bias + ReLU + LayerNorm + (add/max) pooling atomics.
// One wave32 per row (8 rows per 256-thread block); 4 floats per lane.
// Float max done with a monotonic u32 encoding + atomicMax.
// ---------------------------------------------------------------------------
__global__ void ln_pool_kernel(const float* __restrict__ Agg,
                               const float* __restrict__ Hlin,
                               const float* __restrict__ dinv,
                               const float* __restrict__ bgcn,
                               const float* __restrict__ gamma,
                               const float* __restrict__ beta,
                               const int* __restrict__ batch,
                               float* __restrict__ x_add,
                               unsigned* __restrict__ x_max, int n) {
    int wave = threadIdx.x >> 5;
    int lane = threadIdx.x & 31;
    int row  = blockIdx.x * 8 + wave;
    if (row >= n) return;

    float di = dinv[row];
    float selfc = di * di;
    size_t base = (size_t)row * Hh + lane * 4;

    float v[4];
    float s = 0.0f;
    #pragma unroll
    for (int j = 0; j < 4; ++j) {
        float t = Agg[base + j] + Hlin[base + j] * selfc + bgcn[lane * 4 + j];
        t = fmaxf(t, 0.0f);
        v[j] = t;
        s += t;
    }
    #pragma unroll
    for (int off = 16; off > 0; off >>= 1) s += __shfl_xor(s, off, 32);
    float mu = s * (1.0f / Hh);

    float sq = 0.0f;
    #pragma unroll
    for (int j = 0; j < 4; ++j) { float d = v[j] - mu; sq += d * d; }
    #pragma unroll
    for (int off = 16; off > 0; off >>= 1) sq += __shfl_xor(sq, off, 32);
    float rstd = rsqrtf(sq * (1.0f / Hh) + 1e-5f);

    int g = batch[row];
    #pragma unroll
    for (int j = 0; j < 4; ++j) {
        int c = lane * 4 + j;
        float o = (v[j] - mu) * rstd * gamma[c] + beta[c];
        atomicAdd(&x_add[g * Hh + c], o);
        unsigned ub  = __float_as_uint(o);
        unsigned enc = (ub & 0x80000000u) ? ~ub : (ub | 0x80000000u);
        atomicMax(&x_max[g * Hh + c], enc);
    }
}

// ---------------------------------------------------------------------------
// MLP layer 1: g = [mean | add | max] (built on the fly), out = relu(g@W1+b1)
// grid = G, block = 128 (one thread per output column).
// ---------------------------------------------------------------------------
__global__ void mlp1_kernel(const float* __restrict__ x_add,
                            const float* __restrict__ counts,
                            const unsigned* __restrict__ x_max,
                            const float* __restrict__ W1,
                            const float* __restrict__ b1,
                            float* __restrict__ out1) {
    int g = blockIdx.x, j = threadIdx.x;
    float cnt = counts[g];
    float inv = 1.0f / fmaxf(cnt, 1.0f);
    float acc = b1[j];
    for (int k = 0; k < Hh; ++k) {
        float add  = x_add[g * Hh + k];
        float mean = add * inv;
        unsigned e  = x_max[g * Hh + k];
        unsigned ub = (e & 0x80000000u) ? (e & 0x7FFFFFFFu) : ~e;
        float mx = (cnt > 0.0f) ? __uint_as_float(ub) : 0.0f;
        acc += mean * W1[(size_t)k * Hh + j]
             + add  * W1[(size_t)(Hh + k) * Hh + j]
             + mx   * W1[(size_t)(2 * Hh + k) * Hh + j];
    }
    out1[g * Hh + j] = fmaxf(acc, 0.0f);
}

__global__ void mlp2_kernel(const float* __restrict__ in,
                            const float* __restrict__ W2,
                            const float* __restrict__ b2,
                            float* __restrict__ out2) {
    int g = blockIdx.x, j = threadIdx.x;   // j < 64
    float acc = b2[j];
    for (int k = 0; k < Hh; ++k) acc += in[g * Hh + k] * W2[(size_t)k * 64 + j];
    out2[g * 64 + j] = fmaxf(acc, 0.0f);
}

__global__ void mlp3_kernel(const float* __restrict__ in,
                            const float* __restrict__ W3,
                            const float* __restrict__ b3,
                            float* __restrict__ out) {
    int g = blockIdx.x, j = threadIdx.x;   // block 32, use 10
    if (j < 10) {
        float acc = b3[j];
        for (int k = 0; k < 64; ++k) acc += in[g * 64 + k] * W3[k * 10 + j];
        out[g * 10 + j] = acc;
    }
}

// ---------------------------------------------------------------------------
extern "C" void kernel_launch(void* const* d_in, const int* in_sizes, int n_in,
                              void* d_out, int out_size, void* d_ws, size_t ws_size,
                              hipStream_t stream) {
    const float* x      = (const float*)d_in[0];
    const int*   eidx   = (const int*)d_in[1];
    const int*   batch  = (const int*)d_in[2];
    const float* Wg     = (const float*)d_in[4];
    const float* bgcn   = (const float*)d_in[5];
    const float* gamma  = (const float*)d_in[6];
    const float* beta   = (const float*)d_in[7];
    const float* W1     = (const float*)d_in[8];
    const float* b1     = (const float*)d_in[9];
    const float* W2     = (const float*)d_in[10];
    const float* b2     = (const float*)d_in[11];
    const float* W3     = (const float*)d_in[12];
    const float* b3     = (const float*)d_in[13];

    const int N_ = in_sizes[0] / Dd;
    const int E_ = in_sizes[1] / 2;
    const int* srcp = eidx;
    const int* dstp = eidx + E_;

    // workspace layout (floats)
    float*    ws     = (float*)d_ws;
    float*    agg    = ws;                                    // N*H
    float*    deg    = agg + (size_t)N_ * Hh;                 // N  (becomes dinv)
    float*    counts = deg + N_;                              // G
    float*    x_add  = counts + Gg;                           // G*H
    unsigned* x_max  = (unsigned*)(x_add + (size_t)Gg * Hh);  // G*H
    float*    h      = (float*)(x_max + (size_t)Gg * Hh);     // N*H
    float*    out1   = h + (size_t)N_ * Hh;                   // G*H
    float*    out2   = out1 + (size_t)Gg * Hh;                // G*64

    // 1. zero agg, deg, counts, x_add, x_max
    size_t zcnt = (size_t)N_ * Hh + (size_t)N_ + Gg + 2 * (size_t)Gg * Hh;
    zero_ws_kernel<<<(unsigned)((zcnt + 255) / 256), 256, 0, stream>>>(ws, zcnt);

    // 2. h = x @ W_gcn (WMMA f32 16x16x4)
    gemm_xw_wmma<<<(N_ + 15) / 16, 256, 0, stream>>>(x, Wg, h, N_);

    // 3. degree counting
    deg_count_kernel<<<(E_ + 255) / 256, 256, 0, stream>>>(dstp, deg, E_);

    // 4. dinv + graph node counts
    dinv_counts_kernel<<<(N_ + 255) / 256, 256, 0, stream>>>(deg, batch, counts, N_);

    // 5. edge gather/scatter (dominant, memory-bound)
    edge_scatter_kernel<<<(E_ + 7) / 8, 256, 0, stream>>>(srcp, dstp, h, deg, agg, E_);

    // 6. fused self-loop + bias + relu + layernorm + pooling
    ln_pool_kernel<<<(N_ + 7) / 8, 256, 0, stream>>>(agg, h, deg, bgcn, gamma, beta,
                                                     batch, x_add, x_max, N_);

    // 7-9. MLP head
    mlp1_kernel<<<Gg, 128, 0, stream>>>(x_add, counts, x_max, W1, b1, out1);
    mlp2_kernel<<<Gg, 64, 0, stream>>>(out1, W2, b2, out2);
    mlp3_kernel<<<Gg, 32, 0, stream>>>(out2, W3, b3, (float*)d_out);
}